// CausalAttentionSortNet_48747878809990
// MI455X (gfx1250) — compile-verified
//
#include <hip/hip_runtime.h>
#include <math.h>

typedef __attribute__((ext_vector_type(2))) float v2f;
typedef __attribute__((ext_vector_type(8))) float v8f;

#define BH 32
#define NSEQ 8192
#define DIMD 64
#define BUCKET 64
#define NB 128        // query buckets
#define NKV 129       // kv buckets + leading zero pad
#define NPAD 144      // padded to 9 WMMA n-tiles
#define LDS_STRIDE 66 // float stride: even (float2 align) + conflict-reducing pad

// ---------------------------------------------------------------------------
// Stage 1: per (bh, bucket) reductions over q and k. One block per bucket,
// 64 threads = one per feature dim d (coalesced row reads). Bandwidth-bound:
// this kernel streams all 134 MB of q+k exactly once.
//   Bq[b,j,d] = sum_{u} q[b, j*64+u, d]
//   Bk[b,j,d] = sum_{u} k[b, j*64+u, d]
//   Wk[b,j,d] = sum_{u} k[b, j*64+u, d] * T_u,  T_u = sum_{s=u}^{63} 1/(j*64+s+1)
// (backward loop builds the harmonic tail T_u incrementally)
// ---------------------------------------------------------------------------
__global__ void __launch_bounds__(64)
bucket_reduce_kernel(const float* __restrict__ q,
                     const float* __restrict__ k,
                     float* __restrict__ Bq,
                     float* __restrict__ Bk,
                     float* __restrict__ Wk) {
  int b = blockIdx.x >> 7;   // / 128
  int j = blockIdx.x & 127;
  int d = threadIdx.x;       // 0..63
  const size_t base = ((size_t)b * NSEQ + (size_t)j * BUCKET) * DIMD + d;

  float sumq = 0.f, sumk = 0.f, wk = 0.f, t = 0.f;
  for (int u = BUCKET - 1; u >= 0; --u) {
    float kv = k[base + (size_t)u * DIMD];
    float qv = q[base + (size_t)u * DIMD];
    t += 1.0f / (float)(j * BUCKET + u + 1);  // harmonic tail T_u
    sumk += kv;
    wk   += kv * t;
    sumq += qv;
  }
  size_t o = ((size_t)b * NB + j) * DIMD + d;
  Bq[o] = sumq;
  Bk[o] = sumk;
  Wk[o] = wk;
}

// ---------------------------------------------------------------------------
// Stage 2: serial prefix scan over the 128 buckets (tiny: 32 blocks x 64 thr).
//   sq[b,i,d]   = (P_q(i) + q[b, i*64, d]) / (i*64 + 1)
//   sk[b,0,d]   = 0                                  (the jnp.pad row)
//   sk[b,j+1,d] = P_k(j) * H_j + Wk[b,j,d],  H_j = sum_{s=1}^{64} 1/(j*64+s)
// ---------------------------------------------------------------------------
__global__ void __launch_bounds__(64)
scan_kernel(const float* __restrict__ q,
            const float* __restrict__ Bq,
            const float* __restrict__ Bk,
            const float* __restrict__ Wk,
            float* __restrict__ sq,
            float* __restrict__ sk) {
  int b = blockIdx.x;
  int d = threadIdx.x;
  float Pq = 0.f, Pk = 0.f;
  sk[((size_t)b * NKV + 0) * DIMD + d] = 0.f;
  for (int j = 0; j < NB; ++j) {
    float qb = q[((size_t)b * NSEQ + (size_t)j * BUCKET) * DIMD + d];
    sq[((size_t)b * NB + j) * DIMD + d] = (Pq + qb) / (float)(j * BUCKET + 1);
    float H = 0.f;
    for (int s = 1; s <= BUCKET; ++s) H += 1.0f / (float)(j * BUCKET + s);
    size_t o = ((size_t)b * NB + j) * DIMD + d;
    sk[((size_t)b * NKV + j + 1) * DIMD + d] = Pk * H + Wk[o];
    Pq += Bq[o];
    Pk += Bk[o];
  }
}

// ---------------------------------------------------------------------------
// Stage 3: per head, R = (sq @ sk^T) * 8^-1, causal mask, softmax, strict-lower
// filter. One block per head; 8 waves; wave w owns M-tile rows [16w, 16w+16).
// GEMM via V_WMMA_F32_16X16X4_F32, K=64 -> 16 k-steps, N padded 129->144.
// sk staged in LDS (~37 KB) since all 8 waves reuse all 129 rows.
// ---------------------------------------------------------------------------
__global__ void __launch_bounds__(256)
attn_softmax_kernel(const float* __restrict__ sq,
                    const float* __restrict__ sk,
                    float* __restrict__ out) {
  __shared__ float skl[NPAD * LDS_STRIDE];
  int b   = blockIdx.x;
  int tid = threadIdx.x;

  // Stage sk rows 0..128 into LDS, zero the 15 pad rows.
  for (int idx = tid; idx < NPAD * DIMD; idx += 256) {
    int row = idx >> 6, col = idx & 63;
    float v = (row < NKV) ? sk[((size_t)b * NKV + row) * DIMD + col] : 0.f;
    skl[row * LDS_STRIDE + col] = v;
  }
  __syncthreads();

  int wave  = tid >> 5;
  int lane  = tid & 31;
  int l16   = lane & 15;
  int hi    = lane >> 4;        // half-wave select: K offset {0,1} vs {2,3}
  int mbase = wave * 16;
  int m     = mbase + l16;      // A-matrix row for this lane

  const float2* qrow = (const float2*)(sq + ((size_t)b * NB + m) * DIMD);

  v8f c[9];
  #pragma unroll
  for (int nt = 0; nt < 9; ++nt) c[nt] = {};

  // K loop outer -> 9 independent accumulation chains interleave in the pipe.
  #pragma unroll 4
  for (int kk = 0; kk < 16; ++kk) {
    // A 16x4 f32 layout: lane m, VGPRs hold K = {2*hi, 2*hi+1} of k-slab kk.
    float2 af = qrow[2 * kk + hi];
    v2f a; a[0] = af.x; a[1] = af.y;
    #pragma unroll
    for (int nt = 0; nt < 9; ++nt) {
      int ncol = nt * 16 + l16;
      // B 4x16 layout mirrors A per column: B[k][n] = sk[n][k].
      float2 bf = *(const float2*)&skl[ncol * LDS_STRIDE + 4 * kk + 2 * hi];
      v2f bb; bb[0] = bf.x; bb[1] = bf.y;
      c[nt] = __builtin_amdgcn_wmma_f32_16x16x4_f32(
          /*neg_a=*/false, a, /*neg_b=*/false, bb,
          /*c_mod=*/(short)0, c[nt], /*reuse_a=*/false, /*reuse_b=*/false);
    }
  }

  // C/D layout: VGPR r -> row mbase + r + 8*hi, col = ntile*16 + l16.
  const float scale = 0.125f;  // 64^-0.5
  #pragma unroll
  for (int r = 0; r < 8; ++r) {
    int row = mbase + r + 8 * hi;
    float vals[9];
    float mx = -__FLT_MAX__;
    #pragma unroll
    for (int nt = 0; nt < 9; ++nt) {
      int colg = nt * 16 + l16;
      float v = c[nt][r] * scale;
      if (colg > row || colg >= NKV) v = -__FLT_MAX__;  // causal + pad mask
      vals[nt] = v;
      mx = fmaxf(mx, v);
    }
    // row reduction across the 16 lanes of this half-wave
    for (int off = 1; off < 16; off <<= 1)
      mx = fmaxf(mx, __shfl_xor(mx, off, 16));
    float s = 0.f;
    #pragma unroll
    for (int nt = 0; nt < 9; ++nt) {
      float e = __expf(vals[nt] - mx);   // masked -> exp(-huge) = 0
      vals[nt] = e;
      s += e;
    }
    for (int off = 1; off < 16; off <<= 1)
      s += __shfl_xor(s, off, 16);
    float inv = 1.0f / s;
    #pragma unroll
    for (int nt = 0; nt < 9; ++nt) {
      int colg = nt * 16 + l16;
      float p = vals[nt] * inv;
      if (colg >= row) p = 0.f;          // keep strictly-lower only
      if (colg < NKV)
        out[((size_t)b * NB + row) * NKV + colg] = p;
    }
  }
}

// ---------------------------------------------------------------------------
extern "C" void kernel_launch(void* const* d_in, const int* in_sizes, int n_in,
                              void* d_out, int out_size, void* d_ws, size_t ws_size,
                              hipStream_t stream) {
  const float* q = (const float*)d_in[0];
  const float* k = (const float*)d_in[1];
  float* out = (float*)d_out;

  float* ws = (float*)d_ws;
  const size_t nbd = (size_t)BH * NB * DIMD;   // 262144 floats
  float* Bq = ws;
  float* Bk = Bq + nbd;
  float* Wk = Bk + nbd;
  float* sq = Wk + nbd;
  float* sk = sq + nbd;                        // BH*NKV*DIMD floats
  // total workspace: ~5.25 MB

  bucket_reduce_kernel<<<BH * NB, 64, 0, stream>>>(q, k, Bq, Bk, Wk);
  scan_kernel<<<BH, 64, 0, stream>>>(q, Bq, Bk, Wk, sq, sk);
  attn_softmax_kernel<<<BH, 256, 0, stream>>>(sq, sk, out);
}